// IntraGraphAttention_67997922230505
// MI455X (gfx1250) — compile-verified
//
#include <hip/hip_runtime.h>
#include <math.h>

#define B_   4
#define L_   256
#define DN_  256
#define DE_  128
#define H_   8
#define DK_  32
#define DEH_ 16   // DE per head

typedef __attribute__((ext_vector_type(2))) float v2f;
typedef __attribute__((ext_vector_type(8))) float v8f;

__device__ __forceinline__ float lrelu_f(float x) { return x >= 0.f ? x : 0.2f * x; }

// ---------------------------------------------------------------------------
// Generic f32 WMMA GEMM: C = A(MxK) @ Bm(KxN) [+ bias[n]] ; optional epilogue
//   epi==0 : C = acc + bias
//   epi==2 : C = lrelu(acc) + Res   (residual add, used for node_h)
// Batched over blockIdx.z with two-level offsets: z = zo*zInner + zi,
// off = zo*s*_o + zi*s*_i.  One wave computes one 16x16 tile, K-loop step 4.
// ---------------------------------------------------------------------------
__global__ __launch_bounds__(256) void wmma_gemm_f32(
    const float* __restrict__ A, const float* __restrict__ Bm,
    const float* __restrict__ bias, const float* __restrict__ Res,
    float* __restrict__ C,
    int M, int N, int K, int lda, int ldb, int ldc,
    int tiles_n, int zInner,
    long long sA_o, long long sA_i, long long sB_o, long long sB_i,
    long long sC_o, long long sC_i, int epi)
{
    int wid = (blockIdx.x * blockDim.x + threadIdx.x) >> 5;
    int tm = wid / tiles_n;
    int tn = wid - tm * tiles_n;
    int m0 = tm * 16, n0 = tn * 16;
    if (m0 >= M) return;

    int z = blockIdx.z;
    int zo = z / zInner, zi = z - zo * zInner;
    const float* Ap = A + (size_t)(zo * sA_o + zi * sA_i);
    const float* Bp = Bm + (size_t)(zo * sB_o + zi * sB_i);
    long long offC = zo * sC_o + zi * sC_i;

    int lane = threadIdx.x & 31;
    int half = lane >> 4;      // K-half selector for A/B operand layout
    int l16  = lane & 15;

    v8f acc = {};
    size_t arow = (size_t)(m0 + l16) * (size_t)lda;
    for (int k = 0; k < K; k += 4) {
        int ka = k + 2 * half;
        v2f a, b;
        a.x = Ap[arow + ka];
        a.y = Ap[arow + ka + 1];
        b.x = Bp[(size_t)ka * ldb + n0 + l16];
        b.y = Bp[(size_t)(ka + 1) * ldb + n0 + l16];
        acc = __builtin_amdgcn_wmma_f32_16x16x4_f32(
            false, a, false, b, (short)0, acc, false, false);
    }

    float* Cp = C + (size_t)offC;
    int n = n0 + l16;
    if (n < N) {
        float bv = bias ? bias[n] : 0.f;
        const float* Rp = Res ? (Res + (size_t)offC) : nullptr;
        #pragma unroll
        for (int r = 0; r < 8; ++r) {
            int m = m0 + r + 8 * half;
            float v = acc[r] + bv;
            if (epi == 2) v = lrelu_f(v) + Rp[(size_t)m * ldc + n];
            Cp[(size_t)m * ldc + n] = v;
        }
    }
}

// ---------------------------------------------------------------------------
// Fused edge output: new_edge[b,i,j,:] = edge[b,i,j,:] @ Wfused + U[b,i] + V[b,j]
// (diagonal i==j forced to out_e_b). One wave = 16-row strip x all 128 cols
// (8 accumulator tiles) so `edge` streams from HBM exactly once.
// Wfused (64 KB) is staged in LDS once per workgroup, pair-packed along K so
// each B operand pair {Wf[ka][n], Wf[ka+1][n]} is a single ds_load_b64:
// inner loop = 1 global_load_b64 + 8 ds_load_b64 per 8 WMMAs.
// ---------------------------------------------------------------------------
__global__ __launch_bounds__(256) void edge_out_wmma(
    const float* __restrict__ edge, const float* __restrict__ Wf,
    const float* __restrict__ U, const float* __restrict__ V,
    const float* __restrict__ oeb, float* __restrict__ out)
{
    __shared__ float WfL[DE_ * DE_];  // 64 KB of the WGP's 320 KB LDS
    // pack: WfL[((r>>1)*128 + c)*2 + (r&1)] = Wf[r*128 + c]
    for (int idx = threadIdx.x; idx < DE_ * DE_; idx += 256) {
        int r = idx >> 7, c = idx & 127;
        WfL[((((r >> 1) << 7) + c) << 1) + (r & 1)] = Wf[idx];
    }
    __syncthreads();

    int b = blockIdx.y;
    int strip = blockIdx.x * 8 + (threadIdx.x >> 5);  // 0..4095
    int m0 = strip * 16;                              // row within L*L
    int lane = threadIdx.x & 31;
    int half = lane >> 4, l16 = lane & 15;

    const float* Ap = edge + (size_t)b * L_ * L_ * DE_;
    const float2* Wp = (const float2*)WfL;
    v8f acc[8] = {};

    size_t arow = (size_t)(m0 + l16) * DE_;
    __builtin_prefetch(&Ap[arow + 64], 0, 1);  // global_prefetch_b8 on the stream
    for (int k = 0; k < DE_; k += 4) {
        int ka = k + 2 * half;
        int kk = (k >> 1) + half;              // packed K-pair row
        v2f a;
        a.x = Ap[arow + ka];
        a.y = Ap[arow + ka + 1];
        #pragma unroll
        for (int t = 0; t < 8; ++t) {
            float2 w = Wp[(kk << 7) + t * 16 + l16];
            v2f bb;
            bb.x = w.x;
            bb.y = w.y;
            acc[t] = __builtin_amdgcn_wmma_f32_16x16x4_f32(
                false, a, false, bb, (short)0, acc[t], false, false);
        }
    }

    int i = m0 >> 8;  // strip lies entirely within one i (16 | 256)
    const float* Ub = U + ((size_t)b * L_ + i) * DE_;
    const float* Vb = V + (size_t)b * L_ * DE_;
    float* Ob = out + (size_t)b * L_ * L_ * DE_;
    #pragma unroll
    for (int t = 0; t < 8; ++t) {
        int n = t * 16 + l16;
        float uv = Ub[n];
        float db = oeb[n];
        #pragma unroll
        for (int r = 0; r < 8; ++r) {
            int m = m0 + r + 8 * half;
            int j = m & (L_ - 1);
            float v = acc[t][r] + uv + Vb[(size_t)j * DE_ + n];
            if (i == j) v = db;
            Ob[(size_t)m * DE_ + n] = v;
        }
    }
}

// ---------------------------------------------------------------------------
// si/sj: per (b,i,h) dot of node_p head slice with wq / wk
// ---------------------------------------------------------------------------
__global__ void si_sj_kernel(const float* __restrict__ node_p,
                             const float* __restrict__ attn_w,
                             float* __restrict__ si, float* __restrict__ sj)
{
    int t = blockIdx.x * blockDim.x + threadIdx.x;
    if (t >= B_ * L_ * H_) return;
    int h = t & (H_ - 1);
    int bi = t >> 3;
    const float* np = node_p + (size_t)bi * DN_ + h * DK_;
    float a = 0.f, c = 0.f;
    for (int d = 0; d < DK_; ++d) {
        float v = np[d];
        a += v * attn_w[d];
        c += v * attn_w[DK_ + d];
    }
    si[t] = a;
    sj[t] = c;
}

// ---------------------------------------------------------------------------
// Attention softmax with the reference's exact gather/rotation semantics:
//   k=0..254 -> column j=k+(k>=i); probs_pair[i,j] = probs[(j-1) mod 255].
// One wave per (b,h,i) row; writes P[b,h,i,:] (diagonal zeroed).
// ---------------------------------------------------------------------------
__global__ __launch_bounds__(256) void attn_softmax(
    const float* __restrict__ si, const float* __restrict__ sj,
    const float* __restrict__ se, float* __restrict__ P)
{
    int wrow = blockIdx.x * 8 + (threadIdx.x >> 5);  // (b,h,i) flattened
    int b = wrow >> 11;
    int rem = wrow & 2047;
    int h = rem >> 8;
    int i = rem & 255;
    int lane = threadIdx.x & 31;

    float svi = si[((b << 8) + i) * H_ + h];
    float s[8];
    float mx = -1e30f;
    #pragma unroll
    for (int t = 0; t < 8; ++t) {
        int k = lane + 32 * t;
        float v = -1e30f;
        if (k < 255) {
            int j = k + (k >= i ? 1 : 0);
            float sc = svi + sj[((b << 8) + j) * H_ + h]
                     + se[(((size_t)(b << 8) + i) * 256 + j) * H_ + h];
            v = lrelu_f(sc);
        }
        s[t] = v;
        mx = fmaxf(mx, v);
    }
    for (int d = 16; d >= 1; d >>= 1) mx = fmaxf(mx, __shfl_xor(mx, d, 32));
    float sum = 0.f;
    #pragma unroll
    for (int t = 0; t < 8; ++t) {
        float e = (s[t] > -1e29f) ? __expf(s[t] - mx) : 0.f;
        s[t] = e;
        sum += e;
    }
    for (int d = 16; d >= 1; d >>= 1) sum += __shfl_xor(sum, d, 32);
    float inv = 1.f / sum;

    float* Pr = P + (size_t)wrow * 256;
    #pragma unroll
    for (int t = 0; t < 8; ++t) {
        int k = lane + 32 * t;
        if (k < 255) {
            float w = s[t] * inv;
            int jd = k + 1;
            Pr[jd] = (jd == i) ? 0.f : w;
            if (k == 254) Pr[0] = (i == 0) ? 0.f : w;
        }
    }
}

// ---------------------------------------------------------------------------
// Weight-fusion prep kernels (tiny, run once per launch)
// ---------------------------------------------------------------------------
__global__ void prep_ve(const float* We_w, const float* We_b, const float* attn_w,
                        float* VeP, float* ce16)
{
    int t = blockIdx.x * blockDim.x + threadIdx.x;  // 128*16
    if (t >= 128 * 16) return;
    int d = t >> 4, h = t & 15;
    float s = 0.f;
    if (h < H_)
        for (int e = 0; e < DEH_; ++e) s += We_w[d * DE_ + h * DEH_ + e] * attn_w[64 + e];
    VeP[t] = s;
    if (d == 0) {
        float c = 0.f;
        if (h < H_)
            for (int e = 0; e < DEH_; ++e) c += We_b[h * DEH_ + e] * attn_w[64 + e];
        ce16[h] = c;
    }
}

__global__ void prep_bdwh(const float* Wh_w, const float* Wh_b,
                          float* BDWh, float* biasWh)
{
    int t = blockIdx.x * blockDim.x + threadIdx.x;  // 256*256
    if (t >= 256 * 256) return;
    int r = t >> 8, c = t & 255;
    BDWh[t] = ((r >> 5) == (c >> 5)) ? Wh_w[(r & 31) * DK_ + (c & 31)] : 0.f;
    if (r == 0) biasWh[c] = Wh_b[c & 31];
}

__global__ void prep_T(const float* We2_w, const float* We2_b, const float* edge_w,
                       const float* edge_b, float* T, float* cb)
{
    int t = blockIdx.x * blockDim.x + threadIdx.x;  // 128*128
    if (t >= 128 * 128) return;
    int d = t >> 7, c = t & 127;
    int h = c >> 4, e = c & 15;
    float s = 0.f;
    for (int f = 0; f < 16; ++f) s += We2_w[d * DE_ + h * 16 + f] * edge_w[f * 16 + e];
    T[t] = s;
    if (d == 0) {
        float cbv = edge_b[e];
        for (int f = 0; f < 16; ++f) cbv += We2_b[h * 16 + f] * edge_w[f * 16 + e];
        cb[c] = cbv;
    }
}

__global__ void prep_wfused(const float* T, const float* out_e_w, float* Wfused)
{
    int t = blockIdx.x * blockDim.x + threadIdx.x;  // 128*128
    if (t >= 128 * 128) return;
    int d = t >> 7, g = t & 127;
    float s = 0.f;
    for (int c = 0; c < 128; ++c) s += T[d * 128 + c] * out_e_w[c * 128 + g];
    Wfused[t] = s;
}

__global__ void prep_m1(const float* Wn2_w, const float* Wn2_b, const float* edge_w,
                        const float* cb, float* M1i, float* M1j,
                        float* t1i, float* t1j)
{
    int t = blockIdx.x * blockDim.x + threadIdx.x;  // 256*128
    if (t >= 256 * 128) return;
    int d = t >> 7, c = t & 127;
    int h = c >> 4, e = c & 15;
    float a = 0.f, b = 0.f;
    for (int q = 0; q < DK_; ++q) {
        float w = Wn2_w[d * DN_ + h * DK_ + q];
        a += w * edge_w[(16 + q) * 16 + e];
        b += w * edge_w[(48 + q) * 16 + e];
    }
    M1i[t] = a;
    M1j[t] = b;
    if (d == 0) {
        float ti = cb[c], tj = 0.f;
        for (int q = 0; q < DK_; ++q) {
            float wb = Wn2_b[h * DK_ + q];
            ti += wb * edge_w[(16 + q) * 16 + e];
            tj += wb * edge_w[(48 + q) * 16 + e];
        }
        t1i[c] = ti;
        t1j[c] = tj;
    }
}

__global__ void prep_wuv(const float* M1i, const float* M1j, const float* t1i,
                         const float* t1j, const float* out_e_w, const float* out_e_b,
                         float* WU, float* WV, float* bU, float* bV)
{
    int t = blockIdx.x * blockDim.x + threadIdx.x;  // 256*128
    if (t >= 256 * 128) return;
    int d = t >> 7, g = t & 127;
    float su = 0.f, sv = 0.f;
    for (int c = 0; c < 128; ++c) {
        float w = out_e_w[c * 128 + g];
        su += M1i[d * 128 + c] * w;
        sv += M1j[d * 128 + c] * w;
    }
    WU[t] = su;
    WV[t] = sv;
    if (d == 0) {
        float bu = out_e_b[g], bv = 0.f;
        for (int c = 0; c < 128; ++c) {
            float w = out_e_w[c * 128 + g];
            bu += t1i[c] * w;
            bv += t1j[c] * w;
        }
        bU[g] = bu;
        bV[g] = bv;
    }
}

// ---------------------------------------------------------------------------
extern "C" void kernel_launch(void* const* d_in, const int* in_sizes, int n_in,
                              void* d_out, int out_size, void* d_ws, size_t ws_size,
                              hipStream_t stream)
{
    const float* node    = (const float*)d_in[0];
    const float* edge    = (const float*)d_in[1];
    const float* Wn_w    = (const float*)d_in[2];
    const float* Wn_b    = (const float*)d_in[3];
    const float* Wh_w    = (const float*)d_in[4];
    const float* Wh_b    = (const float*)d_in[5];
    const float* We_w    = (const float*)d_in[6];
    const float* We_b    = (const float*)d_in[7];
    const float* Wn2_w   = (const float*)d_in[8];
    const float* Wn2_b   = (const float*)d_in[9];
    const float* We2_w   = (const float*)d_in[10];
    const float* We2_b   = (const float*)d_in[11];
    const float* attn_w  = (const float*)d_in[12];
    const float* edge_w  = (const float*)d_in[13];
    const float* edge_b  = (const float*)d_in[14];
    const float* out_n_w = (const float*)d_in[15];
    const float* out_n_b = (const float*)d_in[16];
    const float* out_e_w = (const float*)d_in[17];
    const float* out_e_b = (const float*)d_in[18];

    float* new_node = (float*)d_out;                       // B*L*DN = 262144
    float* new_edge = (float*)d_out + (size_t)B_ * L_ * DN_;

    float* ws = (float*)d_ws;
    size_t o = 0;
    auto take = [&](size_t n) { float* p = ws + o; o += n; return p; };
    float* se_    = take((size_t)B_ * L_ * L_ * H_);   // 2,097,152
    float* P      = take((size_t)B_ * H_ * L_ * L_);   // 2,097,152
    float* node_p = take((size_t)B_ * L_ * DN_);
    float* Whh    = take((size_t)B_ * L_ * DN_);
    float* node_h = take((size_t)B_ * L_ * DN_);
    float* si_    = take((size_t)B_ * L_ * H_);
    float* sj_    = take((size_t)B_ * L_ * H_);
    float* U      = take((size_t)B_ * L_ * DE_);
    float* V      = take((size_t)B_ * L_ * DE_);
    float* VeP    = take(128 * 16);
    float* ce16   = take(16);
    float* BDWh   = take(256 * 256);
    float* biasWh = take(256);
    float* T      = take(128 * 128);
    float* cb     = take(128);
    float* Wfused = take(128 * 128);
    float* M1i    = take(256 * 128);
    float* M1j    = take(256 * 128);
    float* t1i    = take(128);
    float* t1j    = take(128);
    float* WU     = take(256 * 128);
    float* WV     = take(256 * 128);
    float* bU     = take(128);
    float* bV     = take(128);

    dim3 blk(256);

    // --- weight fusion prep (tiny) ---
    prep_ve<<<8, blk, 0, stream>>>(We_w, We_b, attn_w, VeP, ce16);
    prep_bdwh<<<256, blk, 0, stream>>>(Wh_w, Wh_b, BDWh, biasWh);
    prep_T<<<64, blk, 0, stream>>>(We2_w, We2_b, edge_w, edge_b, T, cb);
    prep_wfused<<<64, blk, 0, stream>>>(T, out_e_w, Wfused);
    prep_m1<<<128, blk, 0, stream>>>(Wn2_w, Wn2_b, edge_w, cb, M1i, M1j, t1i, t1j);
    prep_wuv<<<128, blk, 0, stream>>>(M1i, M1j, t1i, t1j, out_e_w, out_e_b,
                                      WU, WV, bU, bV);

    // --- se = edge @ VeP + ce16  (M=262144, N=8, K=128) ---
    wmma_gemm_f32<<<dim3(2048, 1, 1), blk, 0, stream>>>(
        edge, VeP, ce16, nullptr, se_,
        B_ * L_ * L_, 8, 128, 128, 16, 8, 1, 1, 0, 0, 0, 0, 0, 0, 0);

    // --- node_p = node @ Wn_w + Wn_b ---
    wmma_gemm_f32<<<dim3(128, 1, 1), blk, 0, stream>>>(
        node, Wn_w, Wn_b, nullptr, node_p,
        B_ * L_, DN_, DN_, 256, 256, 256, 16, 1, 0, 0, 0, 0, 0, 0, 0);

    si_sj_kernel<<<32, blk, 0, stream>>>(node_p, attn_w, si_, sj_);

    // --- Whh = node_p @ blockdiag(Wh_w) + Wh_b ---
    wmma_gemm_f32<<<dim3(128, 1, 1), blk, 0, stream>>>(
        node_p, BDWh, biasWh, nullptr, Whh,
        B_ * L_, DN_, DN_, 256, 256, 256, 16, 1, 0, 0, 0, 0, 0, 0, 0);

    attn_softmax<<<1024, blk, 0, stream>>>(si_, sj_, se_, P);

    // --- node_h = node_p + lrelu(P @ Whh)   batched over (b,h): z=32 ---
    wmma_gemm_f32<<<dim3(4, 1, 32), blk, 0, stream>>>(
        P, Whh, nullptr, node_p, node_h,
        L_, DK_, L_, 256, 256, 256, 2, 8,
        (long long)H_ * L_ * L_, (long long)L_ * L_,
        (long long)L_ * DN_, (long long)DK_,
        (long long)L_ * DN_, (long long)DK_, 2);

    // --- new_node = node_h @ out_n_w + out_n_b ---
    wmma_gemm_f32<<<dim3(128, 1, 1), blk, 0, stream>>>(
        node_h, out_n_w, out_n_b, nullptr, new_node,
        B_ * L_, DN_, DN_, 256, 256, 256, 16, 1, 0, 0, 0, 0, 0, 0, 0);

    // --- U = new_node @ WU + bU ; V = new_node @ WV + bV ---
    wmma_gemm_f32<<<dim3(64, 1, 1), blk, 0, stream>>>(
        new_node, WU, bU, nullptr, U,
        B_ * L_, DE_, DN_, 256, 128, 128, 8, 1, 0, 0, 0, 0, 0, 0, 0);
    wmma_gemm_f32<<<dim3(64, 1, 1), blk, 0, stream>>>(
        new_node, WV, bV, nullptr, V,
        B_ * L_, DE_, DN_, 256, 128, 128, 8, 1, 0, 0, 0, 0, 0, 0, 0);

    // --- new_edge = edge @ Wfused + U[b,i] + V[b,j]  (diag -> out_e_b) ---
    edge_out_wmma<<<dim3(512, 4, 1), blk, 0, stream>>>(
        edge, Wfused, U, V, out_e_b, new_edge);
}